// Graph_resnet_58282706206746
// MI455X (gfx1250) — compile-verified
//
#include <hip/hip_runtime.h>
#include <hip/hip_bf16.h>

typedef __attribute__((ext_vector_type(2))) float v2f;
typedef __attribute__((ext_vector_type(8))) float v8f;

#define TB 256

// ---------------------------------------------------------------- utilities
__global__ void zero_int_kernel(int* __restrict__ p, int n) {
  int i = blockIdx.x * blockDim.x + threadIdx.x;
  if (i < n) p[i] = 0;
}

__global__ void copy_int_kernel(const int* __restrict__ a, int* __restrict__ b, int n) {
  int i = blockIdx.x * blockDim.x + threadIdx.x;
  if (i < n) b[i] = a[i];
}

// in-degree histogram (deg = segment_sum(ones, dst))
__global__ void count_kernel(const int* __restrict__ dst, int* __restrict__ cnt, int E) {
  int e = blockIdx.x * blockDim.x + threadIdx.x;
  if (e < E) atomicAdd(&cnt[dst[e]], 1);
}

// single-block two-pass exclusive scan over n counts -> rowptr[0..n]
__global__ void scan_kernel(const int* __restrict__ cnt, int* __restrict__ rowptr, int n) {
  __shared__ int part[1024];
  const int t = threadIdx.x;
  const int chunk = (n + 1023) / 1024;
  const int beg = t * chunk;
  const int end = (beg + chunk < n) ? (beg + chunk) : n;
  int s = 0;
  for (int i = beg; i < end; ++i) s += cnt[i];
  part[t] = s;
  __syncthreads();
  for (int off = 1; off < 1024; off <<= 1) {
    int v = (t >= off) ? part[t - off] : 0;
    __syncthreads();
    part[t] += v;
    __syncthreads();
  }
  int run = (t == 0) ? 0 : part[t - 1];
  for (int i = beg; i < end; ++i) { rowptr[i] = run; run += cnt[i]; }
  if (beg <= n && end == n) rowptr[n] = run;  // all such writers hold the total
}

// dis = deg>0 ? rsqrt(max(deg,1)) : 0   (integer counts: deg>=1 when >0)
__global__ void dis_kernel(const int* __restrict__ cnt, float* __restrict__ dis, int n) {
  int i = blockIdx.x * blockDim.x + threadIdx.x;
  if (i < n) dis[i] = (cnt[i] > 0) ? rsqrtf((float)cnt[i]) : 0.0f;
}

// bucket edges by dst; store src index + sym-normalized weight
__global__ void fill_csr_kernel(const int* __restrict__ src, const int* __restrict__ dst,
                                const float* __restrict__ dis, int* __restrict__ rowfill,
                                int* __restrict__ csrc, float* __restrict__ cw, int E) {
  int e = blockIdx.x * blockDim.x + threadIdx.x;
  if (e < E) {
    int s = src[e], d = dst[e];
    int slot = atomicAdd(&rowfill[d], 1);
    csrc[slot] = s;
    cw[slot] = dis[s] * dis[d];
  }
}

// ------------------------------------------------- pull-mode SpMM (L2-resident)
// tout[v,:] = alpha * sum_{in-edges (u->v)} w * tin[u,:]  (+ beta * tprev[v,:])
// One thread per (node, float4 column group); a wave shares one node for
// F=128 so rowptr/csrc/cw loads broadcast and gathers are contiguous rows.
__global__ void prop_pull_kernel(const float4* __restrict__ tin, float4* __restrict__ tout,
                                 const float4* __restrict__ tprev,
                                 const int* __restrict__ rowptr, const int* __restrict__ csrc,
                                 const float* __restrict__ cw,
                                 float alpha, float beta, int n, int Fq) {
  long tid = (long)blockIdx.x * blockDim.x + threadIdx.x;
  long total = (long)n * Fq;
  if (tid >= total) return;
  int v = (int)(tid / Fq);
  int q = (int)(tid % Fq);
  int beg = rowptr[v], end = rowptr[v + 1];
  float ax = 0.f, ay = 0.f, az = 0.f, aw = 0.f;
  for (int j = beg; j < end; ++j) {
    int u = csrc[j];
    float w = cw[j];
    float4 tv = tin[(long)u * Fq + q];
    ax += w * tv.x; ay += w * tv.y; az += w * tv.z; aw += w * tv.w;
  }
  float4 r;
  if (tprev) {
    float4 p = tprev[tid];
    r.x = alpha * ax + beta * p.x;
    r.y = alpha * ay + beta * p.y;
    r.z = alpha * az + beta * p.z;
    r.w = alpha * aw + beta * p.w;
  } else {
    r.x = alpha * ax; r.y = alpha * ay; r.z = alpha * az; r.w = alpha * aw;
  }
  tout[tid] = r;
}

// ------------------------------------------------------------- WMMA f32 GEMM
// C(M x N) = A(M x F) @ W(F x N) [+ C] [+ bias], N = NT*16 compile-time.
// One wave per 16-row strip; NT accumulators held in VGPRs; K stepped by 4
// with V_WMMA_F32_16X16X4_F32.
// A frag (16x4 f32): lanes 0-15 row=lane K={0,1}; lanes 16-31 row=lane-15? no:
//   row=lane&15, K={2,3} for upper half-wave (ISA 7.12.2).
// B frag (4x16): lane%16 = column, lane-half selects K-half, vgpr j = K within half.
// C/D frag: vgpr r -> row r (+8 for upper half-wave), lane%16 -> column.
template <int NT>
__global__ __launch_bounds__(TB) void wmma_gemm_kernel(
    const float* __restrict__ A, const float* __restrict__ W,
    const float* __restrict__ bias, float* __restrict__ C,
    int M, int F, int accumulate) {
  constexpr int N = NT * 16;
  const int lane = threadIdx.x & 31;
  const int wave = threadIdx.x >> 5;
  const int mTile = blockIdx.x * (TB / 32) + wave;
  if (mTile * 16 >= M) return;           // wave-uniform: EXEC stays all-ones
  const int half = lane >> 4;            // 0: K 0..1 / rows 0..7, 1: K 2..3 / rows 8..15
  const int l16 = lane & 15;
  const int row = mTile * 16 + l16;

  v8f c[NT];
  if (accumulate) {
#pragma unroll
    for (int t = 0; t < NT; ++t)
#pragma unroll
      for (int r = 0; r < 8; ++r)
        c[t][r] = C[(size_t)(mTile * 16 + half * 8 + r) * N + t * 16 + l16];
  } else {
#pragma unroll
    for (int t = 0; t < NT; ++t)
#pragma unroll
      for (int r = 0; r < 8; ++r) c[t][r] = 0.0f;
  }

  const float* arow = A + (size_t)row * F + half * 2;
  for (int k = 0; k < F; k += 4) {
    v2f a;
    a.x = arow[k];
    a.y = arow[k + 1];
    const float* wk = W + (size_t)(k + half * 2) * N;
#pragma unroll
    for (int t = 0; t < NT; ++t) {
      v2f b;
      b.x = wk[t * 16 + l16];
      b.y = wk[N + t * 16 + l16];
      c[t] = __builtin_amdgcn_wmma_f32_16x16x4_f32(
          /*neg_a=*/false, a, /*neg_b=*/false, b,
          /*c_mod=*/(short)0, c[t], /*reuse_a=*/false, /*reuse_b=*/false);
    }
  }

#pragma unroll
  for (int t = 0; t < NT; ++t) {
    float bv = bias ? bias[t * 16 + l16] : 0.0f;
#pragma unroll
    for (int r = 0; r < 8; ++r)
      C[(size_t)(mTile * 16 + half * 8 + r) * N + t * 16 + l16] = c[t][r] + bv;
  }
}

// h_out = relu(acc) + skip
__global__ void relu_add_kernel(float* __restrict__ out, const float* __restrict__ a,
                                const float* __restrict__ s, int total) {
  int i = blockIdx.x * blockDim.x + threadIdx.x;
  if (i < total) out[i] = fmaxf(a[i], 0.0f) + s[i];
}

// hc = concat([h (n x 64), x (n x 128)], axis=1) -> (n x 192)
__global__ void concat_kernel(const float* __restrict__ h, const float* __restrict__ x,
                              float* __restrict__ hc, int n) {
  long i = (long)blockIdx.x * blockDim.x + threadIdx.x;
  if (i >= (long)n * 192) return;
  int row = (int)(i / 192);
  int col = (int)(i % 192);
  hc[i] = (col < 64) ? h[(size_t)row * 64 + col] : x[(size_t)row * 128 + (col - 64)];
}

// ---------------------------------------------------------------- host side
extern "C" void kernel_launch(void* const* d_in, const int* in_sizes, int n_in,
                              void* d_out, int out_size, void* d_ws, size_t ws_size,
                              hipStream_t stream) {
  (void)n_in; (void)out_size; (void)ws_size;
  const float* x = (const float*)d_in[0];
  const int* ei  = (const int*)d_in[1];
  const int n = in_sizes[0] / 128;
  const int E = in_sizes[1] / 2;
  const int* srcp = ei;
  const int* dstp = ei + E;

  const float* W[3]  = {(const float*)d_in[2], (const float*)d_in[6],  (const float*)d_in[10]};
  const float* b[3]  = {(const float*)d_in[3], (const float*)d_in[7],  (const float*)d_in[11]};
  const float* Wsk[3]= {(const float*)d_in[4], (const float*)d_in[8],  (const float*)d_in[12]};
  const float* bsk[3]= {(const float*)d_in[5], (const float*)d_in[9],  (const float*)d_in[13]};
  const float* Wmix = (const float*)d_in[14];
  const float* bmix = (const float*)d_in[15];
  float* out = (float*)d_out;

  // workspace carve-up (256B aligned)
  size_t off = 0;
  auto alloc = [&](size_t bytes) -> char* {
    char* p = (char*)d_ws + off;
    off += (bytes + 255) & ~(size_t)255;
    return p;
  };
  int*   cnt     = (int*)  alloc((size_t)n * 4);
  int*   rowptr  = (int*)  alloc(((size_t)n + 1) * 4);
  int*   rowfill = (int*)  alloc((size_t)n * 4);
  float* dis     = (float*)alloc((size_t)n * 4);
  int*   csrc    = (int*)  alloc((size_t)E * 4);
  float* cw      = (float*)alloc((size_t)E * 4);
  float* t0      = (float*)alloc((size_t)n * 128 * 4);
  float* t1      = (float*)alloc((size_t)n * 192 * 4);   // 192-wide for mix layer
  float* t2      = (float*)alloc((size_t)n * 128 * 4);
  float* accb    = (float*)alloc((size_t)n * 64 * 4);
  float* skipb   = (float*)alloc((size_t)n * 64 * 4);
  float* hB      = (float*)alloc((size_t)n * 64 * 4);
  float* hC      = (float*)alloc((size_t)n * 64 * 4);
  float* hcat    = (float*)alloc((size_t)n * 192 * 4);

  // ---- CSR build (once, reused by all 16 propagation passes) ----
  zero_int_kernel<<<(n + TB - 1) / TB, TB, 0, stream>>>(cnt, n);
  count_kernel<<<(E + TB - 1) / TB, TB, 0, stream>>>(dstp, cnt, E);
  scan_kernel<<<1, 1024, 0, stream>>>(cnt, rowptr, n);
  dis_kernel<<<(n + TB - 1) / TB, TB, 0, stream>>>(cnt, dis, n);
  copy_int_kernel<<<(n + TB - 1) / TB, TB, 0, stream>>>(rowptr, rowfill, n);
  fill_csr_kernel<<<(E + TB - 1) / TB, TB, 0, stream>>>(srcp, dstp, dis, rowfill, csrc, cw, E);

  const int tiles = (n + 15) / 16;
  const int gemm_blocks = (tiles + (TB / 32) - 1) / (TB / 32);

  auto launch_gemm = [&](int nt, const float* A, const float* Wp, const float* biasp,
                         float* Cp, int F, int accflag) {
    if (nt == 4)
      wmma_gemm_kernel<4><<<gemm_blocks, TB, 0, stream>>>(A, Wp, biasp, Cp, n, F, accflag);
    else
      wmma_gemm_kernel<2><<<gemm_blocks, TB, 0, stream>>>(A, Wp, biasp, Cp, n, F, accflag);
  };
  auto launch_pull = [&](const float* tin, float* tout, const float* tprev,
                         float alpha, float beta, int F) {
    long total = (long)n * (F / 4);
    unsigned blocks = (unsigned)((total + TB - 1) / TB);
    prop_pull_kernel<<<blocks, TB, 0, stream>>>(
        (const float4*)tin, (float4*)tout, (const float4*)tprev,
        rowptr, csrc, cw, alpha, beta, n, F / 4);
  };

  // ---- three Kipf blocks: relu(cheb_K6(h)) + cheb_K1_skip(h) ----
  const float* hin = x;
  int F = 128;
  float* houts[3] = {hB, hC, hB};
  for (int l = 0; l < 3; ++l) {
    // out = Tx0 @ W[0] + bias
    launch_gemm(4, hin, W[l], b[l], accb, F, 0);
    // Tx1 = prop(Tx0) = -A_hat Tx0
    launch_pull(hin, t1, nullptr, -1.0f, 0.0f, F);
    launch_gemm(4, t1, W[l] + (size_t)1 * F * 64, nullptr, accb, F, 1);
    // Tx2 = 2*prop(Tx1) - Tx0, fused as alpha=-2, beta=-1
    const float* Tx0p = hin;
    const float* Tx1p = t1;
    float* freep = t0;
    float* otherp = t2;
    for (int k = 2; k < 6; ++k) {
      float* Tx2p = freep;
      launch_pull(Tx1p, Tx2p, Tx0p, -2.0f, -1.0f, F);
      launch_gemm(4, Tx2p, W[l] + (size_t)k * F * 64, nullptr, accb, F, 1);
      freep = (Tx0p == hin) ? otherp : (float*)Tx0p;  // never frees hin
      Tx0p = Tx1p;
      Tx1p = Tx2p;
    }
    // K=1 skip branch + combine
    launch_gemm(4, hin, Wsk[l], bsk[l], skipb, F, 0);
    relu_add_kernel<<<(unsigned)(((long)n * 64 + TB - 1) / TB), TB, 0, stream>>>(
        houts[l], accb, skipb, n * 64);
    hin = houts[l];
    F = 64;
  }

  // ---- mix layer: cheb_K2(concat(h, x)) -> (n x 32) ----
  concat_kernel<<<(unsigned)(((long)n * 192 + TB - 1) / TB), TB, 0, stream>>>(hin, x, hcat, n);
  launch_gemm(2, hcat, Wmix, bmix, out, 192, 0);
  launch_pull(hcat, t1, nullptr, -1.0f, 0.0f, 192);
  launch_gemm(2, t1, Wmix + (size_t)192 * 32, nullptr, out, 192, 1);
}